// Quantizer_9045201125645
// MI455X (gfx1250) — compile-verified
//
#include <hip/hip_runtime.h>
#include <hip/hip_bf16.h>

// Viterbi trellis quantizer, fully fused in LDS per workgroup.
// S=1024 states, NPRED=4, CHUNK=4, T=64 steps, 16 blocks per WG, 256 WGs.
// DP update per 16-state x 16-block tile is one V_WMMA_F32_16X16X4_F32:
//   D = (-2*codebook_tile) @ xt_tile + (minpred + cnorm)   [xnorm dropped: argmin-invariant]

#define NSTATES 1024
#define HIQ     256          // 2^(L-K)
#define TSTEPS  64
#define BT      16           // blocks per workgroup

typedef __attribute__((ext_vector_type(2))) float v2f;
typedef __attribute__((ext_vector_type(8))) float v8f;

__global__ __launch_bounds__(256) void viterbi_quant_kernel(
    const float* __restrict__ arr,          // [1024][1024]
    const float* __restrict__ cb,           // [1024][4]
    float* __restrict__ rec,                // [1024][1024]
    int* __restrict__ states_out)           // [4096][64]  (= [br][bc][T])
{
    // ---- LDS (≈153 KB total; CDNA5 allows up to 320 KB/WG) ----
    __shared__ float    alpha[NSTATES * BT];          // [s][b]   64 KB
    __shared__ float    minpred[HIQ * BT];            // [p][b]   16 KB
    __shared__ unsigned bp[(TSTEPS - 1) * 16 * BT];   // [t][w][b] 2-bit codes, 63 KB
    __shared__ float    cnorm_l[NSTATES];             //          4 KB
    __shared__ float    xtl[4 * BT];                  // [k][b]   256 B
    __shared__ int      statesl[TSTEPS * BT];         // [t][b]   4 KB
    __shared__ float    redv[256];
    __shared__ int      redi[256];

    const int tid  = threadIdx.x;
    const int lane = tid & 31;
    const int wave = tid >> 5;          // 0..7
    const int bcol = lane & 15;         // block column within the 16-wide tile
    const int hi   = lane >> 4;         // half-wave select
    const int k0   = hi * 2;            // K pair owned by this half (A/B f32 K=4 layout)

    const int wgb0 = blockIdx.x * BT;   // first global block index of this WG
    const int br_i = wgb0 >> 6;         // block-row (constant across the 16 blocks)
    const int bc0  = wgb0 & 63;         // first block-col

    // ---- codebook row norms -> LDS ----
    for (int i = tid; i < NSTATES; i += 256) {
        float4 c = ((const float4*)cb)[i];
        cnorm_l[i] = c.x * c.x + c.y * c.y + c.z * c.z + c.w * c.w;
    }

    // ---- stationary A tiles in registers: wave owns states [wave*128, wave*128+128) ----
    // 32-bit A 16x4 layout: lanes 0-15 hold M=lane, K=0(V0),1(V1); lanes 16-31 hold K=2,3.
    v2f Atile[8];
    #pragma unroll
    for (int q = 0; q < 8; ++q) {
        int m = (wave * 8 + q) * 16 + bcol;
        Atile[q][0] = -2.0f * cb[m * 4 + k0];
        Atile[q][1] = -2.0f * cb[m * 4 + k0 + 1];
    }
    __syncthreads();

    // ---- trellis time loop ----
    for (int t = 0; t < TSTEPS; ++t) {
        // stage targets: x[t, 0:4, b] is a contiguous float4 in the source array
        if (tid < 16) {
            int row = br_i * 16 + (t >> 2);
            int col = (bc0 + tid) * 16 + 4 * (t & 3);
            float4 v = *(const float4*)(arr + row * 1024 + col);
            xtl[0 * BT + tid] = v.x; xtl[1 * BT + tid] = v.y;
            xtl[2 * BT + tid] = v.z; xtl[3 * BT + tid] = v.w;
        }

        if (t > 0) {
            // Phase A: min/argmin over the 4 predecessors, shared by groups of 4 states.
            // thread (w = tid>>4, b = tid&15) handles p = w + 16*i, packs 2-bit codes.
            int w = tid >> 4, b = tid & 15;
            unsigned word = 0;
            #pragma unroll 4
            for (int i = 0; i < 16; ++i) {
                int p = w + 16 * i;
                float a0 = alpha[(p      ) * BT + b];
                float a1 = alpha[(p + 256) * BT + b];
                float a2 = alpha[(p + 512) * BT + b];
                float a3 = alpha[(p + 768) * BT + b];
                float m01 = fminf(a0, a1); int j01 = (a1 < a0) ? 1 : 0;   // first-min ties
                float m23 = fminf(a2, a3); int j23 = (a3 < a2) ? 3 : 2;
                float mn  = fminf(m01, m23);
                int   j   = (m23 < m01) ? j23 : j01;
                minpred[p * BT + b] = mn;
                word |= ((unsigned)j) << (2 * i);
            }
            bp[(t - 1) * 256 + w * 16 + b] = word;
        }
        __syncthreads();

        // Phase B: alpha_new tiles via WMMA. B 4x16 f32 layout mirrors A:
        // V0: lanes 0-15 -> K=0 row, lanes 16-31 -> K=2 row; V1: K=1 / K=3.
        v2f Bt;
        Bt[0] = xtl[(k0    ) * BT + bcol];
        Bt[1] = xtl[(k0 + 1) * BT + bcol];
        for (int q = 0; q < 8; ++q) {
            int st = wave * 8 + q;
            v8f C;
            #pragma unroll
            for (int r = 0; r < 8; ++r) {
                int m = st * 16 + r + 8 * hi;             // 16x16 f32 C/D layout
                float c = cnorm_l[m];
                if (t > 0) c += minpred[(m >> 2) * BT + bcol];
                C[r] = c;
            }
            v8f D = __builtin_amdgcn_wmma_f32_16x16x4_f32(
                false, Atile[q], false, Bt, (short)0, C, false, false);
            #pragma unroll
            for (int r = 0; r < 8; ++r) {
                int s = st * 16 + r + 8 * hi;
                alpha[s * BT + bcol] = D[r];
            }
        }
        __syncthreads();
    }

    // ---- terminal argmin over states, per block column ----
    {
        int w = tid >> 4, b = tid & 15;
        float best = 3.402823466e38f; int bests = 0;
        for (int i = 0; i < 64; ++i) {
            int s = w + 16 * i;
            float a = alpha[s * BT + b];
            if (a < best) { best = a; bests = s; }
        }
        redv[tid] = best; redi[tid] = bests;
    }
    __syncthreads();

    // ---- backtrack (16 threads, one per block column) ----
    if (tid < 16) {
        int b = tid;
        float best = redv[b]; int bests = redi[b];
        for (int w = 1; w < 16; ++w) {
            float v = redv[w * 16 + b]; int s = redi[w * 16 + b];
            if (v < best || (v == best && s < bests)) { best = v; bests = s; }
        }
        int state = bests;
        statesl[(TSTEPS - 1) * BT + b] = state;
        for (int t = TSTEPS - 1; t >= 1; --t) {
            int p = state >> 2;
            unsigned word = bp[(t - 1) * 256 + (p & 15) * 16 + b];
            int j = (int)((word >> (2 * (p >> 4))) & 3u);
            state = p + j * HIQ;
            statesl[(t - 1) * BT + b] = state;
        }
    }
    __syncthreads();

    // ---- emit states [block][t] and reconstruction (float4 per (t,b)) ----
    {
        int b  = tid & 15;
        int gb = wgb0 + b;
        #pragma unroll
        for (int i = 0; i < 4; ++i) {
            int t   = (tid >> 4) + 16 * i;
            int stt = statesl[t * BT + b];
            states_out[gb * TSTEPS + t] = stt;
            float4 v = ((const float4*)cb)[stt];
            int row = br_i * 16 + (t >> 2);
            int col = (bc0 + b) * 16 + 4 * (t & 3);
            *(float4*)(rec + row * 1024 + col) = v;
        }
    }
}

extern "C" void kernel_launch(void* const* d_in, const int* in_sizes, int n_in,
                              void* d_out, int out_size, void* d_ws, size_t ws_size,
                              hipStream_t stream) {
    const float* arr = (const float*)d_in[0];   // [1024,1024] f32
    const float* cb  = (const float*)d_in[1];   // [1024,4] f32
    // outputs concatenated: rec (1024*1024 f32) then states (64*64*64 i32)
    float* rec        = (float*)d_out;
    int*   states_out = (int*)((float*)d_out + 1024 * 1024);
    viterbi_quant_kernel<<<256, 256, 0, stream>>>(arr, cb, rec, states_out);
}